// my_loss_49074296324832
// MI455X (gfx1250) — compile-verified
//
#include <hip/hip_runtime.h>
#include <hip/hip_bf16.h>

// Problem constants from the reference
#define BB 64
#define SS 512
#define CC 1000
#define NN (BB * SS)          // 32768 (b,s) pairs

#define BLOCKS 64
#define THREADS 256           // 8 wave32s per block

typedef __attribute__((ext_vector_type(2))) float v2f;
typedef __attribute__((ext_vector_type(8))) float v8f;

// ---------------------------------------------------------------------------
// Kernel 1: scattered gather + -log + hierarchical wave32 reduction.
// Each thread handles NN/(BLOCKS*THREADS) = 2 independent gathers (MLP),
// reduces within its wave via shuffles, waves combine through LDS.
// One f32 partial per block written to d_ws. Fully deterministic order.
// ---------------------------------------------------------------------------
__global__ void nll_gather_partial(const float* __restrict__ probs,
                                   const int* __restrict__ tgt,
                                   float* __restrict__ partial) {
    const int tid    = blockIdx.x * blockDim.x + threadIdx.x;
    const int stride = BLOCKS * THREADS;

    float acc = 0.0f;
    // NN is a multiple of stride (32768 / 16384 = 2): no tail divergence,
    // EXEC stays all-ones through the gather loop.
    #pragma unroll
    for (int i = tid; i < NN; i += stride) {
        const int t = tgt[i];                       // coalesced int32 load
        const float p = probs[(size_t)i * CC + t];  // scattered gather
        acc -= __logf(p);                           // v_log_f32 path
    }

    // wave32 tree reduction
    #pragma unroll
    for (int off = 16; off > 0; off >>= 1)
        acc += __shfl_down(acc, off, 32);

    __shared__ float smem[THREADS / 32];
    const int lane = threadIdx.x & 31;
    const int wave = threadIdx.x >> 5;
    if (lane == 0) smem[wave] = acc;
    __syncthreads();

    if (wave == 0) {
        float v = (lane < (THREADS / 32)) ? smem[lane] : 0.0f;
        #pragma unroll
        for (int off = 4; off > 0; off >>= 1)
            v += __shfl_down(v, off, 32);
        if (lane == 0) partial[blockIdx.x] = v;
    }
}

// ---------------------------------------------------------------------------
// Kernel 2: finalize 64 partials with the matrix engine, exactly in f32.
// A (16x4 f32) holds the 64 partials, B = all-ones, C = 0:
//   D[m][n] = sum_k A[m][k]  (f32 accumulate — no precision loss)
// One wave; EXEC all-ones as WMMA requires. A-layout per ISA §7.12.2:
//   lane L: VGPR0 = A[L%16][2*(L/16)], VGPR1 = A[L%16][2*(L/16)+1]
// C/D layout: c[v] on lanes 0-15 = row v, on lanes 16-31 = row v+8.
// ---------------------------------------------------------------------------
__global__ void nll_finalize_wmma(const float* __restrict__ partial,
                                  float* __restrict__ out) {
    const int lane  = threadIdx.x;        // 0..31
    const int m     = lane & 15;
    const int kbase = (lane >> 4) << 1;   // 0 or 2

    v2f a;
    a.x = partial[m * 4 + kbase];
    a.y = partial[m * 4 + kbase + 1];

    v2f b;
    b.x = 1.0f;
    b.y = 1.0f;

    v8f c = {};
    c = __builtin_amdgcn_wmma_f32_16x16x4_f32(
            /*neg_a=*/false, a, /*neg_b=*/false, b,
            /*c_mod=*/(short)0, c, /*reuse_a=*/false, /*reuse_b=*/false);

    // lanes 0-15: sum of rowsums 0..7 ; lanes 16-31: sum of rowsums 8..15
    float s = c[0] + c[1] + c[2] + c[3] + c[4] + c[5] + c[6] + c[7];
    s += __shfl_down(s, 16, 32);          // fold upper half-wave
    if (lane == 0) out[0] = s;
}

// ---------------------------------------------------------------------------
extern "C" void kernel_launch(void* const* d_in, const int* in_sizes, int n_in,
                              void* d_out, int out_size, void* d_ws, size_t ws_size,
                              hipStream_t stream) {
    const float* probs = (const float*)d_in[0]; // [B,S,C] float32
    const int*   tgt   = (const int*)d_in[1];   // [B,S]   int32
    float*       dst   = (float*)d_out;         // scalar
    float*       part  = (float*)d_ws;          // 64 floats of scratch

    nll_gather_partial<<<BLOCKS, THREADS, 0, stream>>>(probs, tgt, part);
    nll_finalize_wmma<<<1, 32, 0, stream>>>(part, dst);
}